// EdgeDecoder_52630529245697
// MI455X (gfx1250) — compile-verified
//
#include <hip/hip_runtime.h>

// EdgeDecoder: out[e] = relu(z_drug[row[e]]·W1d + z_effect[col[e]]·W1e + b1) · W2 + b2
// H = 128. Compute-bound (~65.8 GFLOP fp32); all operand data is L2-resident
// on MI455X (z_drug 51MB + z_effect 26MB + idx 8MB << 192MB L2), so the win is
// running the GEMMs on the WMMA pipe with exact fp32 semantics:
// V_WMMA_F32_16X16X4_F32, 16-edge tiles, W1 staged in LDS pre-swizzled into
// B-fragment layout, A-fragments as direct per-lane float2 gathers (L2).
//
// LDS layout interleaves the drug/effect weight halves per K-step so all 16
// fragments of one K-step live in a 4KB window -> immediate DS offsets.

typedef float v2f __attribute__((ext_vector_type(2)));
typedef float v8f __attribute__((ext_vector_type(8)));

#define HDIM   128
#define KSTEPS 32   // HDIM / 4  (K per WMMA = 4)
#define NTILES 8    // HDIM / 16 (N per WMMA = 16)

__global__ __launch_bounds__(256, 1)
void edge_decoder_wmma_f32(const float* __restrict__ z_drug,
                           const float* __restrict__ z_effect,
                           const int*   __restrict__ eli,     // (2, nEdges)
                           const float* __restrict__ W1,      // (256, 128)
                           const float* __restrict__ b1,      // (128)
                           const float* __restrict__ W2,      // (128, 1)
                           const float* __restrict__ b2,      // (1)
                           float*       __restrict__ out,     // (nEdges)
                           int nEdges, int nTiles)
{
    // sW[kk][mat][nt][lane] : mat 0 = W1d (rows 0..127), mat 1 = W1e (rows 128..255)
    // fragment value = { W[kk*4 + (lane>>4)*2 + 0][nt*16 + (lane&15)],
    //                    W[kk*4 + (lane>>4)*2 + 1][nt*16 + (lane&15)] }
    __shared__ v2f   sW[KSTEPS * 2 * NTILES * 32];   // 128 KB
    __shared__ float sB1[HDIM];
    __shared__ float sW2[HDIM];

    const int tid = threadIdx.x;

    // ---- Stage W1 (both halves) into LDS, pre-swizzled to WMMA B layout ----
    for (int i = tid; i < KSTEPS * 2 * NTILES * 32; i += 256) {
        const int kk   = i >> 9;          // / (2*NTILES*32)
        const int mat  = (i >> 8) & 1;
        const int nt   = (i >> 5) & 7;
        const int ln   = i & 31;
        const int krow = kk * 4 + ((ln >> 4) << 1) + mat * HDIM;
        const int col  = nt * 16 + (ln & 15);
        v2f w;
        w.x = W1[(krow    ) * HDIM + col];
        w.y = W1[(krow + 1) * HDIM + col];
        sW[i] = w;
    }
    if (tid < HDIM) { sB1[tid] = b1[tid]; sW2[tid] = W2[tid]; }
    __syncthreads();

    const int lane = tid & 31;
    const int wave = tid >> 5;
    const int lm   = lane & 15;   // edge-in-tile (A rows are M = lane&15)
    const int lh   = lane >> 4;   // half-wave selects K sub-pair {0,1} vs {2,3}
    const float b2v = b2[0];

    for (int tile = blockIdx.x * 8 + wave; tile < nTiles; tile += gridDim.x * 8) {
        const int e0 = tile * 16;
        int eIdx = e0 + lm;
        if (eIdx >= nEdges) eIdx = nEdges - 1;          // clamp tail (dup reads)
        const int rowId = eli[eIdx];
        const int colId = eli[nEdges + eIdx];
        const float* __restrict__ ad = z_drug   + (size_t)rowId * HDIM + lh * 2;
        const float* __restrict__ ae = z_effect + (size_t)colId * HDIM + lh * 2;

        v8f acc[NTILES] = {};   // zero-initialized accumulators (64 VGPRs)

        #pragma unroll 2
        for (int kk = 0; kk < KSTEPS; ++kk) {
            // A-fragments: per-lane float2 == exact 16x4 f32 WMMA A layout
            const v2f a_d = *(const v2f*)(ad + kk * 4);
            const v2f a_e = *(const v2f*)(ae + kk * 4);
            // all 16 B-fragments for this kk within one 4KB immediate window
            const v2f* __restrict__ bk = &sW[(kk * 2 * NTILES) * 32 + lane];
            #pragma unroll
            for (int n = 0; n < NTILES; ++n) {
                acc[n] = __builtin_amdgcn_wmma_f32_16x16x4_f32(
                    false, a_d, false, bk[n * 32], (short)0, acc[n], false, false);
                acc[n] = __builtin_amdgcn_wmma_f32_16x16x4_f32(
                    false, a_e, false, bk[(NTILES + n) * 32], (short)0, acc[n], false, false);
            }
        }

        // ---- Epilogue: +b1, ReLU, ·W2, reduce over N (16 lanes per half) ----
        // C/D layout: VGPR v, lane -> element (M = lh*8 + v, N = lm)
        float partial[8];
        #pragma unroll
        for (int v = 0; v < 8; ++v) partial[v] = 0.f;

        #pragma unroll
        for (int n = 0; n < NTILES; ++n) {
            const float bv = sB1[n * 16 + lm];
            const float wv = sW2[n * 16 + lm];
            #pragma unroll
            for (int v = 0; v < 8; ++v) {
                float hv = acc[n][v] + bv;
                hv = hv > 0.f ? hv : 0.f;
                partial[v] += hv * wv;
            }
        }
        #pragma unroll
        for (int v = 0; v < 8; ++v) {
            float s = partial[v];
            s += __shfl_xor(s, 1, 32);
            s += __shfl_xor(s, 2, 32);
            s += __shfl_xor(s, 4, 32);
            s += __shfl_xor(s, 8, 32);
            partial[v] = s;
        }
        if (lm == 0) {
            #pragma unroll
            for (int v = 0; v < 8; ++v) {
                const int e = e0 + lh * 8 + v;
                if (e < nEdges) out[e] = partial[v] + b2v;
            }
        }
    }
}

extern "C" void kernel_launch(void* const* d_in, const int* in_sizes, int n_in,
                              void* d_out, int out_size, void* d_ws, size_t ws_size,
                              hipStream_t stream) {
    (void)n_in; (void)out_size; (void)d_ws; (void)ws_size;
    const float* z_drug   = (const float*)d_in[0];
    const float* z_effect = (const float*)d_in[1];
    const int*   eli      = (const int*)  d_in[2];
    const float* W1       = (const float*)d_in[3];
    const float* b1       = (const float*)d_in[4];
    const float* W2       = (const float*)d_in[5];
    const float* b2       = (const float*)d_in[6];
    float* out = (float*)d_out;

    const int nEdges = in_sizes[2] / 2;
    const int nTiles = (nEdges + 15) / 16;

    int blocks = 1024;                       // persistent-ish, grid-stride
    const int maxBlocks = (nTiles + 7) / 8;  // 8 waves (tiles) per block pass
    if (blocks > maxBlocks) blocks = maxBlocks;

    edge_decoder_wmma_f32<<<blocks, 256, 0, stream>>>(
        z_drug, z_effect, eli, W1, b1, W2, b2, out, nEdges, nTiles);
}